// DecFCSwitch_18863496364234
// MI455X (gfx1250) — compile-verified
//
#include <hip/hip_runtime.h>

// ---------------------------------------------------------------------------
// DecFCSwitch (MoE-style switch FC) for MI455X / gfx1250.
//
// Route tokens by y_index, compute only the selected branch per token with
// V_WMMA_F32_16X16X4_F32 (full fp32 precision). 32-token tiles: each weight
// fragment feeds two M-row WMMA tiles -> 0.5 global loads per WMMA.
// ---------------------------------------------------------------------------

typedef __attribute__((ext_vector_type(2))) float v2f;
typedef __attribute__((ext_vector_type(8))) float v8f;

#define B_TOK   4096
#define NBR     16
#define S_DIM   256
#define D_DIM   1024
#define TILE_M  32
#define KCHUNK  128
#define MAX_TILES (B_TOK / TILE_M + NBR)   // 144: worst-case padded tiles

// workspace layout (int words)
#define WS_COUNTS 0                         // [16]
#define WS_OFFS   16                        // [17] padded segment offsets
#define WS_FILL   40                        // [16]
#define WS_NTILES 60                        // [1]
#define WS_TEXP   64                        // [MAX_TILES] expert per tile
#define WS_TSTART (64 + MAX_TILES)          // [MAX_TILES] tok-list start per tile
#define WS_TOK    (64 + 2 * MAX_TILES)      // [B_TOK + NBR*TILE_M] grouped tokens (-1 pad)
#define WS_TOTAL  (WS_TOK + B_TOK + NBR * TILE_M)

// D = A(16x4) * B(4x16) + C, fp32 (codegen-confirmed on gfx1250).
__device__ __forceinline__ v8f wmma4(v2f a, v2f b, v8f c) {
  return __builtin_amdgcn_wmma_f32_16x16x4_f32(
      /*neg_a=*/false, a, /*neg_b=*/false, b,
      /*c_mod=*/(short)0, c, /*reuse_a=*/false, /*reuse_b=*/false);
}

// ---------------- routing ----------------
__global__ void ws_init_kernel(int* __restrict__ ws) {
  int i = blockIdx.x * blockDim.x + threadIdx.x;
  if (i < WS_TOTAL) ws[i] = (i >= WS_TOK) ? -1 : 0;
}

__global__ void ws_count_kernel(const long long* __restrict__ yidx, int* __restrict__ ws) {
  int b = blockIdx.x * blockDim.x + threadIdx.x;
  if (b < B_TOK) {
    int e = (int)yidx[b];
    atomicAdd(&ws[WS_COUNTS + e], 1);
  }
}

__global__ void ws_scan_kernel(int* __restrict__ ws) {
  if (threadIdx.x == 0 && blockIdx.x == 0) {
    int off = 0, idx = 0;
    for (int e = 0; e < NBR; ++e) {
      ws[WS_OFFS + e] = off;
      int cnt   = ws[WS_COUNTS + e];
      int tiles = (cnt + TILE_M - 1) / TILE_M;
      for (int t = 0; t < tiles; ++t) {
        ws[WS_TEXP + idx]   = e;
        ws[WS_TSTART + idx] = off + t * TILE_M;
        ++idx;
      }
      off += tiles * TILE_M;
    }
    ws[WS_OFFS + NBR] = off;
    ws[WS_NTILES]     = idx;
  }
}

__global__ void ws_scatter_kernel(const long long* __restrict__ yidx, int* __restrict__ ws) {
  int b = blockIdx.x * blockDim.x + threadIdx.x;
  if (b < B_TOK) {
    int e   = (int)yidx[b];
    int pos = ws[WS_OFFS + e] + atomicAdd(&ws[WS_FILL + e], 1);
    ws[WS_TOK + pos] = b;
  }
}

// ---------------- fused dual-GEMM per 32-token tile ----------------
// LDS: A chunk 32x128 f32 (stride 132) + hid 32x256 f32 (stride 260) ~ 50 KB.
// Both strides == 4 (mod 64 banks) -> conflict-free b64 fragment loads.
__launch_bounds__(256)
__global__ void moe_gemm_kernel(const float* __restrict__ x,
                                const float* __restrict__ W_in,
                                const float* __restrict__ b_in,
                                const float* __restrict__ W_out,
                                const float* __restrict__ b_out,
                                const int* __restrict__ ws,
                                float* __restrict__ out) {
  __shared__ float Alds[TILE_M][KCHUNK + 4];
  __shared__ float Hlds[TILE_M][S_DIM + 4];
  __shared__ int   toks[TILE_M];

  const int bid = blockIdx.x;
  if (bid >= ws[WS_NTILES]) return;            // wave-uniform: EXEC stays full
  const int e      = ws[WS_TEXP + bid];
  const int tstart = ws[WS_TSTART + bid];

  const int tid  = threadIdx.x;
  const int lane = tid & 31;
  const int wave = tid >> 5;                   // 0..7
  const int nrow = lane & 15;                  // row/col index within 16-wide frag
  const int half = lane >> 4;                  // K-pair select (K0K1 vs K2K3)

  if (tid < TILE_M) toks[tid] = ws[WS_TOK + tstart + tid];
  __syncthreads();

  const float* We = W_in + (size_t)e * S_DIM * D_DIM;
  const float* Wo = W_out + (size_t)e * D_DIM * S_DIM;
  __builtin_prefetch(Wo + (size_t)(wave * 128) * S_DIM, 0, 1);  // warm GEMM2 stream

  // -------- GEMM1: hid[32x256] = relu(x[toks]) @ W_in[e]^T, K chunked -------
  v8f z = {0.f, 0.f, 0.f, 0.f, 0.f, 0.f, 0.f, 0.f};
  v8f acc00 = z, acc01 = z, acc10 = z, acc11 = z;   // [ntile][mtile]
  const int nt0 = wave * 2, nt1 = wave * 2 + 1;

  for (int kc = 0; kc < D_DIM; kc += KCHUNK) {
    {   // cooperative stage: 32 rows x 128 cols, relu applied (8 thr/row)
      const int row = tid >> 3;
      const int c0  = (tid & 7) * 16;
      const int t   = toks[row];
      const float4* src =
          (const float4*)(x + (size_t)(t < 0 ? 0 : t) * D_DIM + kc + c0);
#pragma unroll
      for (int q = 0; q < 4; ++q) {
        float4 v = (t >= 0) ? src[q] : make_float4(0.f, 0.f, 0.f, 0.f);
        v.x = fmaxf(v.x, 0.f); v.y = fmaxf(v.y, 0.f);
        v.z = fmaxf(v.z, 0.f); v.w = fmaxf(v.w, 0.f);
        *(float4*)&Alds[row][c0 + 4 * q] = v;
      }
    }
    __syncthreads();
#pragma unroll 4
    for (int k = 0; k < KCHUNK; k += 4) {
      v2f a0 = *(const v2f*)&Alds[nrow][k + 2 * half];
      v2f a1 = *(const v2f*)&Alds[16 + nrow][k + 2 * half];
      v2f b0 = *(const v2f*)(We + (size_t)(nt0 * 16 + nrow) * D_DIM + kc + k + 2 * half);
      v2f b1 = *(const v2f*)(We + (size_t)(nt1 * 16 + nrow) * D_DIM + kc + k + 2 * half);
      acc00 = wmma4(a0, b0, acc00);
      acc01 = wmma4(a1, b0, acc01);
      acc10 = wmma4(a0, b1, acc10);
      acc11 = wmma4(a1, b1, acc11);
    }
    __syncthreads();
  }

  {   // hid(+bias) -> LDS.  C/D layout: M = v + 8*half (per 16-row tile)
    float bi0 = b_in[(size_t)e * S_DIM + nt0 * 16 + nrow];
    float bi1 = b_in[(size_t)e * S_DIM + nt1 * 16 + nrow];
#pragma unroll
    for (int v = 0; v < 8; ++v) {
      int m = v + 8 * half;
      Hlds[m][nt0 * 16 + nrow]      = acc00[v] + bi0;
      Hlds[16 + m][nt0 * 16 + nrow] = acc01[v] + bi0;
      Hlds[m][nt1 * 16 + nrow]      = acc10[v] + bi1;
      Hlds[16 + m][nt1 * 16 + nrow] = acc11[v] + bi1;
    }
  }
  __syncthreads();

  // -------- GEMM2: out[32x1024] = hid @ W_out[e]^T, K = 256, two passes ----
  for (int pass = 0; pass < 2; ++pass) {
    v8f acc[4][2];
#pragma unroll
    for (int j = 0; j < 4; ++j) { acc[j][0] = z; acc[j][1] = z; }
    const int ntb = wave * 8 + pass * 4;       // first of 4 N-tiles this pass

    for (int k = 0; k < S_DIM; k += 4) {
      v2f a0 = *(const v2f*)&Hlds[nrow][k + 2 * half];
      v2f a1 = *(const v2f*)&Hlds[16 + nrow][k + 2 * half];
#pragma unroll
      for (int j = 0; j < 4; ++j) {
        int d  = (ntb + j) * 16 + nrow;
        v2f bf = *(const v2f*)(Wo + (size_t)d * S_DIM + k + 2 * half);
        acc[j][0] = wmma4(a0, bf, acc[j][0]);
        acc[j][1] = wmma4(a1, bf, acc[j][1]);
      }
    }

    // epilogue: out = x + branch_out + b_out, guarded for pad tokens
#pragma unroll
    for (int j = 0; j < 4; ++j) {
      int   d  = (ntb + j) * 16 + nrow;
      float bo = b_out[(size_t)e * D_DIM + d];
#pragma unroll
      for (int mt = 0; mt < 2; ++mt) {
#pragma unroll
        for (int v = 0; v < 8; ++v) {
          int m = mt * 16 + v + 8 * half;
          int t = toks[m];
          if (t >= 0) {
            size_t off = (size_t)t * D_DIM + d;
            out[off] = x[off] + acc[j][mt][v] + bo;
          }
        }
      }
    }
  }
}

// ---------------------------------------------------------------------------
extern "C" void kernel_launch(void* const* d_in, const int* in_sizes, int n_in,
                              void* d_out, int out_size, void* d_ws, size_t ws_size,
                              hipStream_t stream) {
  const float*     x     = (const float*)d_in[0];
  const long long* yidx  = (const long long*)d_in[1];   // int64 per reference
  const float*     W_in  = (const float*)d_in[2];
  const float*     b_in  = (const float*)d_in[3];
  const float*     W_out = (const float*)d_in[4];
  const float*     b_out = (const float*)d_in[5];
  float*           out   = (float*)d_out;
  int*             ws    = (int*)d_ws;

  ws_init_kernel<<<(WS_TOTAL + 255) / 256, 256, 0, stream>>>(ws);
  ws_count_kernel<<<B_TOK / 256, 256, 0, stream>>>(yidx, ws);
  ws_scan_kernel<<<1, 32, 0, stream>>>(ws);
  ws_scatter_kernel<<<B_TOK / 256, 256, 0, stream>>>(yidx, ws);
  moe_gemm_kernel<<<MAX_TILES, 256, 0, stream>>>(x, W_in, b_in, W_out, b_out, ws, out);
}